// GraphConvolution_49074296324789
// MI455X (gfx1250) — compile-verified
//
#include <hip/hip_runtime.h>

typedef float v2f __attribute__((ext_vector_type(2)));
typedef float v8f __attribute__((ext_vector_type(8)));

#define N_NODES 100000
#define N_EDGES 1600000
#define IN_F 256
#define OUT_F 64

// ---------------------------------------------------------------------------
// Kernel 1: out[n, f] = b[f]   (also absorbs the "+ b" of the reference)
// ---------------------------------------------------------------------------
__global__ void __launch_bounds__(256) gcn_init_out(float* __restrict__ out,
                                                    const float* __restrict__ b) {
    int i = blockIdx.x * 256 + threadIdx.x;
    out[i] = b[i & (OUT_F - 1)];
}

// ---------------------------------------------------------------------------
// Kernel 2: support[N, 64] = x[N, 256] @ W[256, 64] using V_WMMA_F32_16X16X4_F32
// Block = 256 threads = 8 waves; each wave computes one 16x16 tile.
// Wave w: row-tile = w/4, col-tile = w%4  -> block covers 32 rows x 64 cols.
// 100000 / 32 = 3125 blocks exactly -> no divergence, EXEC all ones for WMMA.
//
// LDS holds the whole weight in layout [k/4][n][k%4] so that the B-matrix
// operand pair (W[k+koff][n], W[k+koff+1][n]) per lane is one aligned b64 load.
// ---------------------------------------------------------------------------
__global__ void __launch_bounds__(256) gcn_gemm(const float* __restrict__ x,
                                                const float* __restrict__ w,
                                                float* __restrict__ support) {
    __shared__ float ldsW[IN_F * OUT_F];   // 64 KB

    const int tid = threadIdx.x;
    // Cooperative preload: coalesced global reads, scattered LDS writes.
    for (int i = tid; i < IN_F * OUT_F; i += 256) {
        int k = i >> 6;                    // 0..255
        int n = i & 63;                    // 0..63
        ldsW[((k >> 2) << 8) + (n << 2) + (k & 3)] = w[i];
    }
    __syncthreads();

    const int lane    = tid & 31;
    const int wv      = tid >> 5;                       // 0..7
    const int rowbase = blockIdx.x * 32 + (wv >> 2) * 16;
    const int ncol    = (wv & 3) * 16 + (lane & 15);    // global output column
    const int koff    = (lane >> 4) << 1;               // 0 or 2 (A/B K pair)
    const int m       = rowbase + (lane & 15);          // A-matrix row for this lane

    const float* xrow = x + (size_t)m * IN_F;

    v8f acc = {0.f, 0.f, 0.f, 0.f, 0.f, 0.f, 0.f, 0.f};

#pragma unroll 8
    for (int k = 0; k < IN_F; k += 4) {
        // A 16x4 fp32 tile: lane holds x[m][k+koff], x[m][k+koff+1] (8B aligned)
        v2f a = *(const v2f*)(xrow + k + koff);
        // B 4x16 fp32 tile: lane holds W[k+koff][ncol], W[k+koff+1][ncol]
        v2f bv = *(const v2f*)(&ldsW[((k >> 2) << 8) + (ncol << 2) + koff]);
        // D = A x B + C  (8 args: neg_a, A, neg_b, B, c_mod, C, reuse_a, reuse_b)
        acc = __builtin_amdgcn_wmma_f32_16x16x4_f32(false, a, false, bv,
                                                    (short)0, acc, false, false);
    }

    // C/D layout: VGPR v -> M = v (lanes 0-15) or v+8 (lanes 16-31); N = lane&15
    const int mout = rowbase + ((lane >> 4) << 3);
    float* dst = support + (size_t)mout * OUT_F + ncol;
#pragma unroll
    for (int v = 0; v < 8; ++v)
        dst[(size_t)v * OUT_F] = acc[v];
}

// ---------------------------------------------------------------------------
// Kernel 3: edge-parallel SpMM. One wave per edge; lane owns 2 features.
// support (25.6 MB) is resident in the 192 MB L2 -> gathers + atomics stay
// on-chip. 1600000 / 8 = 200000 blocks exactly.
// ---------------------------------------------------------------------------
__global__ void __launch_bounds__(256) gcn_spmm(const float* __restrict__ support,
                                                const int* __restrict__ rows,
                                                const int* __restrict__ cols,
                                                const float* __restrict__ vals,
                                                float* __restrict__ out) {
    const int lane = threadIdx.x & 31;
    const int e    = blockIdx.x * 8 + (threadIdx.x >> 5);

    const int   r = rows[e];   // same address across wave -> broadcast load
    const int   c = cols[e];
    const float v = vals[e];

    const int f = lane << 1;   // feature pair
    v2f s = *(const v2f*)(support + (size_t)c * OUT_F + f);

    float* o = out + (size_t)r * OUT_F + f;
    atomicAdd(o,     v * s.x);
    atomicAdd(o + 1, v * s.y);
}

// ---------------------------------------------------------------------------
extern "C" void kernel_launch(void* const* d_in, const int* in_sizes, int n_in,
                              void* d_out, int out_size, void* d_ws, size_t ws_size,
                              hipStream_t stream) {
    const float* x        = (const float*)d_in[0];
    const int*   adj_row  = (const int*)  d_in[1];
    const int*   adj_col  = (const int*)  d_in[2];
    const float* adj_vals = (const float*)d_in[3];
    const float* weight   = (const float*)d_in[4];
    const float* bvec     = (const float*)d_in[5];

    float* out     = (float*)d_out;
    float* support = (float*)d_ws;     // 100000*64*4 = 25.6 MB scratch

    gcn_init_out<<<(N_NODES * OUT_F) / 256, 256, 0, stream>>>(out, bvec);
    gcn_gemm   <<<N_NODES / 32,            256, 0, stream>>>(x, weight, support);
    gcn_spmm   <<<N_EDGES / 8,             256, 0, stream>>>(support, adj_row,
                                                             adj_col, adj_vals, out);
}